// GATModule_53850299957842
// MI455X (gfx1250) — compile-verified
//
#include <hip/hip_runtime.h>
#include <math.h>

// ---------------------------------------------------------------------------
// GATv2 hetero conv for MI455X (gfx1250, wave32).
// Only the last layer's weights matter (reference overwrites outputs per
// layer and never updates x_dict), so we compute exactly one layer.
// Dense x@W goes through V_WMMA_F32_16X16X4_F32 (fp32-faithful).
// ---------------------------------------------------------------------------

typedef float v2f __attribute__((ext_vector_type(2)));
typedef float v8f __attribute__((ext_vector_type(8)));

#define D_HID 128
#define NHEAD 4
#define NEG_SLOPE 0.2f
#define ROWTILES 16  // 16x16-row tiles per block => 256 rows/block

// ------------------------- order-preserving float<->uint key ----------------
__device__ __forceinline__ unsigned fkey(float f) {
  unsigned u = __float_as_uint(f);
  return (u & 0x80000000u) ? ~u : (u | 0x80000000u);
}
__device__ __forceinline__ float funkey(unsigned u) {
  u = (u & 0x80000000u) ? (u & 0x7FFFFFFFu) : ~u;
  return __uint_as_float(u);
}
#define FKEY_NEG_INF 0x007FFFFFu  // fkey(-inf)

// ------------------------- GEMM: Y[N x 128] = X[N x 128] * W[128 x 128] -----
// Block: 256 threads = 8 waves. Wave w owns the 16-column tile [16w,16w+16)
// and loops over ROWTILES row tiles, reusing a register-resident B panel
// (128x16 slice of W = 64 VGPRs/lane) across all row tiles. Both the preload
// loop and the K loop are FULLY unrolled so breg[] indices are constants and
// the panel stays in VGPRs (a rolled loop would force a scratch spill).
//
// WMMA f32 16x16x4 register layouts (ISA 7.12.2):
//   A: lanes 0-15 hold M=lane with K={0,1}; lanes 16-31 hold M=lane-16,K={2,3}
//   B: v0 -> K=0 (lanes 0-15) / K=2 (lanes 16-31); v1 -> K=1 / K=3
//   D: VGPR i -> M = i + 8*(lane/16), N = lane%16
__global__ __launch_bounds__(256) void gemm128_wmma(
    const float* __restrict__ X, const float* __restrict__ W,
    float* __restrict__ Y, int N) {
  const int lane  = threadIdx.x & 31;
  const int wave  = threadIdx.x >> 5;          // 0..7 -> 16-col tile
  const int col0  = wave << 4;
  const int l16   = lane & 15;
  const int khalf = (lane >> 4) << 1;          // 0 or 2
  const int n     = col0 + l16;

  // ---- preload B panel for this column tile (held in VGPRs) ----
  v2f breg[32];
#pragma unroll
  for (int kk = 0; kk < 32; ++kk) {
    const int k = (kk << 2) + khalf;
    breg[kk].x = W[k * D_HID + n];
    breg[kk].y = W[(k + 1) * D_HID + n];
  }

  const int rowBase = blockIdx.x * (ROWTILES << 4);
  for (int t = 0; t < ROWTILES; ++t) {
    const int row0 = rowBase + (t << 4);
    if (row0 >= N) return;

    int m = row0 + l16;
    if (m >= N) m = N - 1;                     // safe clamp (ragged tail only)
    const float* __restrict__ xrow = X + (size_t)m * D_HID;

    v8f acc = {};
#pragma unroll
    for (int kk = 0; kk < 32; ++kk) {          // FULL unroll: breg stays in VGPRs
      v2f a = *(const v2f*)(xrow + (kk << 2) + khalf);
      acc = __builtin_amdgcn_wmma_f32_16x16x4_f32(
          /*neg_a=*/false, a, /*neg_b=*/false, breg[kk],
          /*c_mod=*/(short)0, acc, /*reuse_a=*/false, /*reuse_b=*/false);
    }

    const int orow = row0 + ((lane >> 4) << 3);
    float* __restrict__ yp = Y + (size_t)orow * D_HID + n;
    if (orow + 7 < N) {                        // full tile: straight-line path
#pragma unroll
      for (int i = 0; i < 8; ++i) yp[(size_t)i * D_HID] = acc[i];
    } else {
#pragma unroll
      for (int i = 0; i < 8; ++i)
        if (orow + i < N) yp[(size_t)i * D_HID] = acc[i];
    }
  }
}

// ------------------------- init kernels -------------------------------------
__global__ void init_seg(unsigned* __restrict__ segmax,
                         float* __restrict__ segsum, int n) {
  int i = blockIdx.x * blockDim.x + threadIdx.x;
  if (i < n) { segmax[i] = FKEY_NEG_INF; segsum[i] = 0.0f; }
}

__global__ void init_out(float* __restrict__ out,
                         const float* __restrict__ bias, long long n) {
  long long i = (long long)blockIdx.x * blockDim.x + threadIdx.x;
  if (i < n) out[i] = bias[i & (D_HID - 1)];
}

// ------------------------- pass 1: per-edge scores + segment max -------------
// One wave per edge. Lane handles channels 4*lane..4*lane+3, which all live in
// head (lane>>3); 8-lane xor-shuffle reduction yields 4 head scores per wave.
__global__ __launch_bounds__(256) void edge_scores(
    const float* __restrict__ yl, const float* __restrict__ yr,
    const int* __restrict__ src, const int* __restrict__ dst, int E,
    const float* __restrict__ att, float* __restrict__ esc,
    unsigned* __restrict__ segmax) {
  const int gw = (int)((blockIdx.x * 256 + threadIdx.x) >> 5);
  const int lane = threadIdx.x & 31;
  if (gw >= E) return;
  const int s = src[gw], d = dst[gw];

  const float4 xl = *(const float4*)(yl + (size_t)s * D_HID + lane * 4);
  const float4 xr = *(const float4*)(yr + (size_t)d * D_HID + lane * 4);
  const float4 a  = *(const float4*)(att + lane * 4);

  float p = 0.0f, z;
  z = xl.x + xr.x; z = (z > 0.0f) ? z : z * NEG_SLOPE; p += a.x * z;
  z = xl.y + xr.y; z = (z > 0.0f) ? z : z * NEG_SLOPE; p += a.y * z;
  z = xl.z + xr.z; z = (z > 0.0f) ? z : z * NEG_SLOPE; p += a.z * z;
  z = xl.w + xr.w; z = (z > 0.0f) ? z : z * NEG_SLOPE; p += a.w * z;

  p += __shfl_xor(p, 1, 32);
  p += __shfl_xor(p, 2, 32);
  p += __shfl_xor(p, 4, 32);

  if ((lane & 7) == 0) {
    const int h = lane >> 3;
    esc[(size_t)gw * NHEAD + h] = p;
    atomicMax(&segmax[(size_t)d * NHEAD + h], fkey(p));
  }
}

// ------------------------- pass 2: exp + segment sum -------------------------
// One thread per (edge, head). Overwrites the score buffer with exp weights.
__global__ __launch_bounds__(256) void edge_expsum(
    const int* __restrict__ dst, int E, float* __restrict__ esc,
    const unsigned* __restrict__ segmax, float* __restrict__ segsum) {
  const int idx = blockIdx.x * blockDim.x + threadIdx.x;
  if (idx >= E * NHEAD) return;
  const int e = idx >> 2, h = idx & 3;
  const int d = dst[e];
  const float m = funkey(segmax[(size_t)d * NHEAD + h]);
  const float w = expf(esc[idx] - m);
  esc[idx] = w;
  atomicAdd(&segsum[(size_t)d * NHEAD + h], w);
}

// ------------------------- pass 3: weighted scatter --------------------------
// One wave per edge; lane adds 4 channels of alpha_h * yl[src] into out[dst].
__global__ __launch_bounds__(256) void edge_scatter(
    const float* __restrict__ yl, const int* __restrict__ src,
    const int* __restrict__ dst, int E, const float* __restrict__ w,
    const float* __restrict__ segsum, float* __restrict__ out) {
  const int gw = (int)((blockIdx.x * 256 + threadIdx.x) >> 5);
  const int lane = threadIdx.x & 31;
  if (gw >= E) return;
  const int s = src[gw], d = dst[gw];
  const int h = lane >> 3;
  const float denom = fmaxf(segsum[(size_t)d * NHEAD + h], 1e-16f);
  const float alpha = w[(size_t)gw * NHEAD + h] / denom;

  const float4 v = *(const float4*)(yl + (size_t)s * D_HID + lane * 4);
  float* o = out + (size_t)d * D_HID + lane * 4;
  atomicAdd(o + 0, alpha * v.x);
  atomicAdd(o + 1, alpha * v.y);
  atomicAdd(o + 2, alpha * v.z);
  atomicAdd(o + 3, alpha * v.w);
}

// ---------------------------------------------------------------------------
extern "C" void kernel_launch(void* const* d_in, const int* in_sizes, int n_in,
                              void* d_out, int out_size, void* d_ws,
                              size_t ws_size, hipStream_t stream) {
  (void)n_in; (void)out_size; (void)ws_size;

  const float* x_asset  = (const float*)d_in[0];
  const float* x_market = (const float*)d_in[1];
  const int*   edge_aa  = (const int*)d_in[2];
  const int*   edge_am  = (const int*)d_in[3];
  const float* Wl       = (const float*)d_in[4];
  const float* Wr       = (const float*)d_in[5];
  const float* att      = (const float*)d_in[6];
  const float* bias     = (const float*)d_in[7];

  const int Na  = in_sizes[0] / D_HID;
  const int Nm  = in_sizes[1] / D_HID;
  const int Eaa = in_sizes[2] / 2;
  const int Eam = in_sizes[3] / 2;
  const int L   = in_sizes[4] / (D_HID * D_HID);

  // Only the last layer's result survives in the reference.
  const float* Wl1   = Wl + (size_t)(L - 1) * D_HID * D_HID;
  const float* Wr1   = Wr + (size_t)(L - 1) * D_HID * D_HID;
  const float* att1  = att + (size_t)(L - 1) * NHEAD * 32;
  const float* bias1 = bias + (size_t)(L - 1) * D_HID;

  const int* src_aa = edge_aa;
  const int* dst_aa = edge_aa + Eaa;
  const int* src_am = edge_am;
  const int* dst_am = edge_am + Eam;

  // ---- carve workspace (256B aligned slabs) ----
  char* ws = (char*)d_ws;
  size_t off = 0;
  auto carve = [&](size_t bytes) {
    char* p = ws + off;
    off += (bytes + 255) & ~(size_t)255;
    return p;
  };
  float*    yl_a     = (float*)carve((size_t)Na * D_HID * 4);   // x_asset @ Wl
  float*    yr_a     = (float*)carve((size_t)Na * D_HID * 4);   // x_asset @ Wr
  float*    yr_m     = (float*)carve((size_t)Nm * D_HID * 4);   // x_market @ Wr
  float*    esc_aa   = (float*)carve((size_t)Eaa * NHEAD * 4);  // scores -> exp w
  float*    esc_am   = (float*)carve((size_t)Eam * NHEAD * 4);
  unsigned* segmax_a = (unsigned*)carve((size_t)Na * NHEAD * 4);
  float*    segsum_a = (float*)carve((size_t)Na * NHEAD * 4);
  unsigned* segmax_m = (unsigned*)carve((size_t)Nm * NHEAD * 4);
  float*    segsum_m = (float*)carve((size_t)Nm * NHEAD * 4);

  float* out_asset  = (float*)d_out;
  float* out_market = out_asset + (size_t)Na * D_HID;

  // ---- dense transforms (WMMA), 256 rows per block ----
  const int rowsPerBlock = ROWTILES * 16;
  gemm128_wmma<<<(Na + rowsPerBlock - 1) / rowsPerBlock, 256, 0, stream>>>(
      x_asset, Wl1, yl_a, Na);
  gemm128_wmma<<<(Na + rowsPerBlock - 1) / rowsPerBlock, 256, 0, stream>>>(
      x_asset, Wr1, yr_a, Na);
  gemm128_wmma<<<(Nm + rowsPerBlock - 1) / rowsPerBlock, 256, 0, stream>>>(
      x_market, Wr1, yr_m, Nm);

  // ---- init segment state + outputs (bias broadcast) ----
  init_seg<<<(Na * NHEAD + 255) / 256, 256, 0, stream>>>(segmax_a, segsum_a,
                                                         Na * NHEAD);
  init_seg<<<(Nm * NHEAD + 255) / 256, 256, 0, stream>>>(segmax_m, segsum_m,
                                                         Nm * NHEAD);
  {
    long long na_elems = (long long)Na * D_HID;
    long long nm_elems = (long long)Nm * D_HID;
    init_out<<<(unsigned)((na_elems + 255) / 256), 256, 0, stream>>>(
        out_asset, bias1, na_elems);
    init_out<<<(unsigned)((nm_elems + 255) / 256), 256, 0, stream>>>(
        out_market, bias1, nm_elems);
  }

  // ---- edge phase ----
  edge_scores<<<(Eaa + 7) / 8, 256, 0, stream>>>(yl_a, yr_a, src_aa, dst_aa,
                                                 Eaa, att1, esc_aa, segmax_a);
  edge_scores<<<(Eam + 7) / 8, 256, 0, stream>>>(yl_a, yr_m, src_am, dst_am,
                                                 Eam, att1, esc_am, segmax_m);

  edge_expsum<<<(Eaa * NHEAD + 255) / 256, 256, 0, stream>>>(
      dst_aa, Eaa, esc_aa, segmax_a, segsum_a);
  edge_expsum<<<(Eam * NHEAD + 255) / 256, 256, 0, stream>>>(
      dst_am, Eam, esc_am, segmax_m, segsum_m);

  edge_scatter<<<(Eaa + 7) / 8, 256, 0, stream>>>(yl_a, src_aa, dst_aa, Eaa,
                                                  esc_aa, segsum_a, out_asset);
  edge_scatter<<<(Eam + 7) / 8, 256, 0, stream>>>(yl_a, src_am, dst_am, Eam,
                                                  esc_am, segsum_m, out_market);
}